// TinyGIN_21251498181384
// MI455X (gfx1250) — compile-verified
//
#include <hip/hip_runtime.h>

typedef __attribute__((ext_vector_type(2))) float v2f;
typedef __attribute__((ext_vector_type(8))) float v8f;

#define N_NODES 100000
#define N_EDGES 3200000
#define HID     32
#define NLAYERS 4
#define NGRAPH  1000
#define NCLS    2
#define BN_EPS  1e-5f

// gfx1250 async-copy-to-LDS path (ASYNCcnt-tracked). The builtin takes
// (int4 as1* gsrc, int4 as3* ldst, imm offset, imm cpol) per the probe.
#if defined(__AMDGCN__) && __has_builtin(__builtin_amdgcn_global_load_async_to_lds_b128) && __has_builtin(__builtin_amdgcn_s_wait_asynccnt)
#define GIN_ASYNC_LDS 1
typedef int gin_i4 __attribute__((ext_vector_type(4)));
typedef __attribute__((address_space(1))) gin_i4* gin_g4p;
typedef __attribute__((address_space(3))) gin_i4* gin_l4p;
#else
#define GIN_ASYNC_LDS 0
#endif

// ---------------- encoder: h = x @ W_enc + b_enc  (F_IN = 1) ----------------
__global__ void gin_enc_kernel(const float* __restrict__ x,
                               const float* __restrict__ W_enc,
                               const float* __restrict__ b_enc,
                               float* __restrict__ h) {
  int i4 = blockIdx.x * blockDim.x + threadIdx.x;
  if (i4 >= N_NODES * (HID / 4)) return;
  int node = i4 >> 3;
  int c0 = (i4 & 7) * 4;
  float xv = x[node];
  float4 o;
  o.x = fmaf(xv, W_enc[c0 + 0], b_enc[c0 + 0]);
  o.y = fmaf(xv, W_enc[c0 + 1], b_enc[c0 + 1]);
  o.z = fmaf(xv, W_enc[c0 + 2], b_enc[c0 + 2]);
  o.w = fmaf(xv, W_enc[c0 + 3], b_enc[c0 + 3]);
  ((float4*)h)[i4] = o;
}

// ---------------- edge scatter: agg[dst] += h[src] ----------------
// 8 threads per edge; each thread gathers 4 channels with one b128 load and
// issues 4 f32 atomics. h (12.8MB) lives in the 192MB L2, so both the gather
// and the atomics resolve at L2 -- HBM only sees the edge indices.
__global__ void gin_scatter_kernel(const float* __restrict__ h,
                                   const int* __restrict__ src,
                                   const int* __restrict__ dst,
                                   float* __restrict__ agg) {
  long long tid = (long long)blockIdx.x * blockDim.x + threadIdx.x;
  if (tid >= (long long)N_EDGES * 8) return;
  int e  = (int)(tid >> 3);
  int c4 = (int)(tid & 7);
  int s = src[e];
  int d = dst[e];
  float4 v = ((const float4*)h)[(long long)s * (HID / 4) + c4];
  float* ap = &agg[(long long)d * HID + c4 * 4];
  atomicAdd(ap + 0, v.x);
  atomicAdd(ap + 1, v.y);
  atomicAdd(ap + 2, v.z);
  atomicAdd(ap + 3, v.w);
}

// ---------------- fused GIN MLP: z=(1+eps)h+agg; z2=relu(z@W1+b1)@W2+b2 -----
// 256 threads = 8 waves; block covers 128 nodes; each wave owns a 16-row tile.
// W1/W2 async-copied to LDS so the GEMM inner loop is pure ds_load +
// v_wmma_f32_16x16x4_f32 with zero global traffic.
__global__ __launch_bounds__(256)
void gin_mlp_kernel(const float* __restrict__ h,
                    const float* __restrict__ agg,
                    const float* __restrict__ epsArr, int layer,
                    const float* __restrict__ W1, const float* __restrict__ b1,
                    const float* __restrict__ W2, const float* __restrict__ b2,
                    float* __restrict__ z2out,
                    float* __restrict__ bnstats /* [0,32)=sum [32,64)=sumsq */) {
  __shared__ float zS[128 * HID];
  __shared__ float w1S[HID * HID];
  __shared__ float w2S[HID * HID];
  __shared__ float bnS[2 * HID];

  const int tid = threadIdx.x;
  const int blockBase = blockIdx.x * 128;
  const float epsv = 1.0f + epsArr[layer];

  if (tid < 2 * HID) bnS[tid] = 0.0f;

  // ---- stage weights into LDS (one async b128 per thread per matrix) ----
#if GIN_ASYNC_LDS
  __builtin_amdgcn_global_load_async_to_lds_b128(
      (gin_g4p)(W1 + tid * 4), (gin_l4p)(w1S + tid * 4), 0, 0);
  __builtin_amdgcn_global_load_async_to_lds_b128(
      (gin_g4p)(W2 + tid * 4), (gin_l4p)(w2S + tid * 4), 0, 0);
#else
  for (int i = tid; i < HID * HID; i += 256) {
    w1S[i] = W1[i];
    w2S[i] = W2[i];
  }
#endif

  // ---- stage z = (1+eps)*h + agg for 128 nodes (float4, zero-pad tail) ----
  {
    const float4* h4 = (const float4*)h;
    const float4* a4 = (const float4*)agg;
    float4* z4 = (float4*)zS;
    for (int i4 = tid; i4 < 128 * (HID / 4); i4 += 256) {
      int gnode = blockBase + (i4 >> 3);
      float4 v;
      if (gnode < N_NODES) {
        long long gi = (long long)gnode * (HID / 4) + (i4 & 7);
        float4 hv = h4[gi];
        float4 av = a4[gi];
        v.x = fmaf(epsv, hv.x, av.x);
        v.y = fmaf(epsv, hv.y, av.y);
        v.z = fmaf(epsv, hv.z, av.z);
        v.w = fmaf(epsv, hv.w, av.w);
      } else {
        v = make_float4(0.f, 0.f, 0.f, 0.f);
      }
      z4[i4] = v;
    }
  }
#if GIN_ASYNC_LDS
  __builtin_amdgcn_s_wait_asynccnt(0);
#endif
  __syncthreads();

  const int wave = tid >> 5;
  const int lane = tid & 31;
  const int l    = lane & 15;             // fragment column / row-within-16
  const int hi2  = (lane >= 16) ? 2 : 0;  // K offset split per ISA A/B layout
  const int rowBase = wave * 16;

  // ---- GEMM1: z @ W1 + b1, then ReLU ----
  v8f c0, c1;
  {
    const float bv0 = b1[l];
    const float bv1 = b1[16 + l];
    for (int r = 0; r < 8; ++r) { c0[r] = bv0; c1[r] = bv1; }
    for (int kk = 0; kk < HID; kk += 4) {
      v2f a;
      a.x = zS[(rowBase + l) * HID + kk + hi2];
      a.y = zS[(rowBase + l) * HID + kk + hi2 + 1];
      v2f b;
      b.x = w1S[(kk + hi2) * HID + l];
      b.y = w1S[(kk + hi2 + 1) * HID + l];
      c0 = __builtin_amdgcn_wmma_f32_16x16x4_f32(false, a, false, b, (short)0, c0, false, false);
      v2f bb;
      bb.x = w1S[(kk + hi2) * HID + 16 + l];
      bb.y = w1S[(kk + hi2 + 1) * HID + 16 + l];
      c1 = __builtin_amdgcn_wmma_f32_16x16x4_f32(false, a, false, bb, (short)0, c1, false, false);
    }
    for (int r = 0; r < 8; ++r) {
      c0[r] = fmaxf(c0[r], 0.0f);
      c1[r] = fmaxf(c1[r], 0.0f);
    }
  }

  // write z1 back into this wave's own 16-row LDS slice (per-wave LDS is
  // in-order, and no other wave touches these rows -> no block barrier needed)
  {
    const int mofs = (lane >= 16) ? 8 : 0;
    for (int r = 0; r < 8; ++r) {
      zS[(rowBase + r + mofs) * HID + l]      = c0[r];
      zS[(rowBase + r + mofs) * HID + 16 + l] = c1[r];
    }
  }

  // ---- GEMM2: z1 @ W2 + b2 ----
  v8f d0, d1;
  {
    const float bv0 = b2[l];
    const float bv1 = b2[16 + l];
    for (int r = 0; r < 8; ++r) { d0[r] = bv0; d1[r] = bv1; }
    for (int kk = 0; kk < HID; kk += 4) {
      v2f a;
      a.x = zS[(rowBase + l) * HID + kk + hi2];
      a.y = zS[(rowBase + l) * HID + kk + hi2 + 1];
      v2f b;
      b.x = w2S[(kk + hi2) * HID + l];
      b.y = w2S[(kk + hi2 + 1) * HID + l];
      d0 = __builtin_amdgcn_wmma_f32_16x16x4_f32(false, a, false, b, (short)0, d0, false, false);
      v2f bb;
      bb.x = w2S[(kk + hi2) * HID + 16 + l];
      bb.y = w2S[(kk + hi2 + 1) * HID + 16 + l];
      d1 = __builtin_amdgcn_wmma_f32_16x16x4_f32(false, a, false, bb, (short)0, d1, false, false);
    }
  }

  // ---- store z2 + BatchNorm partial sums (masked to valid nodes) ----
  {
    const int mofs = (lane >= 16) ? 8 : 0;
    float s0 = 0.f, q0 = 0.f, s1 = 0.f, q1 = 0.f;
    for (int r = 0; r < 8; ++r) {
      int node = blockBase + rowBase + r + mofs;
      if (node < N_NODES) {
        z2out[(long long)node * HID + l]      = d0[r];
        z2out[(long long)node * HID + 16 + l] = d1[r];
        s0 += d0[r]; q0 += d0[r] * d0[r];
        s1 += d1[r]; q1 += d1[r] * d1[r];
      }
    }
    atomicAdd(&bnS[l],            s0);
    atomicAdd(&bnS[HID + l],      q0);
    atomicAdd(&bnS[16 + l],       s1);
    atomicAdd(&bnS[HID + 16 + l], q1);
  }
  __syncthreads();
  if (tid < 2 * HID) atomicAdd(&bnstats[tid], bnS[tid]);
}

// ---------------- BN prep: fold mu/var/gamma/beta into scale+shift ----------
__global__ void gin_bn_prep_kernel(float* __restrict__ bnstats,
                                   const float* __restrict__ gamma,
                                   const float* __restrict__ beta) {
  int c = threadIdx.x;
  if (c >= HID) return;
  const float invN = 1.0f / (float)N_NODES;
  float mu  = bnstats[c] * invN;
  float var = bnstats[HID + c] * invN - mu * mu;
  float rs  = rsqrtf(var + BN_EPS);
  float sc  = gamma[c] * rs;
  bnstats[2 * HID + c] = sc;                 // scale
  bnstats[3 * HID + c] = beta[c] - mu * sc;  // shift
}

// ---------------- BN apply + ReLU: h = relu(z2 * scale + shift) ------------
__global__ void gin_bn_relu_kernel(const float* __restrict__ z2,
                                   const float* __restrict__ bnstats,
                                   float* __restrict__ h) {
  int i4 = blockIdx.x * blockDim.x + threadIdx.x;
  if (i4 >= N_NODES * (HID / 4)) return;
  int c0 = (i4 & 7) * 4;
  const float* sc = bnstats + 2 * HID;
  const float* sh = bnstats + 3 * HID;
  float4 z = ((const float4*)z2)[i4];
  float4 o;
  o.x = fmaxf(fmaf(z.x, sc[c0 + 0], sh[c0 + 0]), 0.0f);
  o.y = fmaxf(fmaf(z.y, sc[c0 + 1], sh[c0 + 1]), 0.0f);
  o.z = fmaxf(fmaf(z.z, sc[c0 + 2], sh[c0 + 2]), 0.0f);
  o.w = fmaxf(fmaf(z.w, sc[c0 + 3], sh[c0 + 3]), 0.0f);
  ((float4*)h)[i4] = o;
}

// ---------------- global add pool ----------------
__global__ void gin_pool_kernel(const float* __restrict__ h,
                                const int* __restrict__ batch,
                                float* __restrict__ pooled) {
  int i4 = blockIdx.x * blockDim.x + threadIdx.x;
  if (i4 >= N_NODES * (HID / 4)) return;
  int node = i4 >> 3;
  int c4 = i4 & 7;
  int g = batch[node];
  float4 v = ((const float4*)h)[i4];
  float* pp = &pooled[g * HID + c4 * 4];
  atomicAdd(pp + 0, v.x);
  atomicAdd(pp + 1, v.y);
  atomicAdd(pp + 2, v.z);
  atomicAdd(pp + 3, v.w);
}

// ---------------- classifier: out = pooled @ W_cls + b_cls ----------------
__global__ void gin_cls_kernel(const float* __restrict__ pooled,
                               const float* __restrict__ W_cls,
                               const float* __restrict__ b_cls,
                               float* __restrict__ out) {
  int idx = blockIdx.x * blockDim.x + threadIdx.x;
  if (idx >= NGRAPH * NCLS) return;
  int g = idx / NCLS, c = idx % NCLS;
  float acc = b_cls[c];
  for (int k = 0; k < HID; ++k)
    acc += pooled[g * HID + k] * W_cls[k * NCLS + c];
  out[idx] = acc;
}

extern "C" void kernel_launch(void* const* d_in, const int* in_sizes, int n_in,
                              void* d_out, int out_size, void* d_ws, size_t ws_size,
                              hipStream_t stream) {
  const float* x     = (const float*)d_in[0];
  const int*   ei    = (const int*)  d_in[1];   // [2, E]
  const int*   batch = (const int*)  d_in[2];
  const float* W_enc = (const float*)d_in[3];
  const float* b_enc = (const float*)d_in[4];
  const float* eps   = (const float*)d_in[5];
  const float* W1    = (const float*)d_in[6];
  const float* b1    = (const float*)d_in[7];
  const float* W2    = (const float*)d_in[8];
  const float* b2    = (const float*)d_in[9];
  const float* gamma = (const float*)d_in[10];
  const float* beta  = (const float*)d_in[11];
  const float* W_cls = (const float*)d_in[12];
  const float* b_cls = (const float*)d_in[13];
  float* out = (float*)d_out;

  const int* srcIdx = ei;
  const int* dstIdx = ei + N_EDGES;

  const size_t featBytes = (size_t)N_NODES * HID * sizeof(float);
  char* ws = (char*)d_ws;
  float* h       = (float*)(ws);
  float* agg     = (float*)(ws + featBytes);
  float* z2      = (float*)(ws + 2 * featBytes);
  float* pooled  = (float*)(ws + 3 * featBytes);
  float* bnstats = (float*)(ws + 3 * featBytes + (size_t)NGRAPH * HID * sizeof(float));

  const int elem4Blocks   = (N_NODES * (HID / 4) + 255) / 256;              // 3125
  const int scatterBlocks = (int)(((long long)N_EDGES * 8 + 255) / 256);    // 100000
  const int mlpBlocks     = (N_NODES + 127) / 128;                          // 782

  gin_enc_kernel<<<elem4Blocks, 256, 0, stream>>>(x, W_enc, b_enc, h);

  for (int l = 0; l < NLAYERS; ++l) {
    (void)hipMemsetAsync(agg, 0, featBytes, stream);
    (void)hipMemsetAsync(bnstats, 0, 2 * HID * sizeof(float), stream);
    gin_scatter_kernel<<<scatterBlocks, 256, 0, stream>>>(h, srcIdx, dstIdx, agg);
    gin_mlp_kernel<<<mlpBlocks, 256, 0, stream>>>(h, agg, eps, l,
                                                  W1 + l * HID * HID, b1 + l * HID,
                                                  W2 + l * HID * HID, b2 + l * HID,
                                                  z2, bnstats);
    gin_bn_prep_kernel<<<1, 32, 0, stream>>>(bnstats, gamma + l * HID, beta + l * HID);
    gin_bn_relu_kernel<<<elem4Blocks, 256, 0, stream>>>(z2, bnstats, h);
  }

  (void)hipMemsetAsync(pooled, 0, (size_t)NGRAPH * HID * sizeof(float), stream);
  gin_pool_kernel<<<elem4Blocks, 256, 0, stream>>>(h, batch, pooled);
  gin_cls_kernel<<<(NGRAPH * NCLS + 255) / 256, 256, 0, stream>>>(pooled, W_cls, b_cls, out);
}